// GMMActivation2DFullyLearnable_65343632441447
// MI455X (gfx1250) — compile-verified
//
#include <hip/hip_runtime.h>
#include <hip/hip_bf16.h>
#include <stdint.h>

#define TPB   256   // 8 waves (wave32)
#define DEPTH 8     // async pipeline stages = tiles per block, 8 x 4KiB = 32 KiB LDS

// ---- gfx1250 async global->LDS feature detection (device pass only) ----
#if defined(__HIP_DEVICE_COMPILE__) && defined(__gfx1250__) && \
    __has_builtin(__builtin_amdgcn_global_load_async_to_lds_b128)
#define ASYNC_OK 1
#else
#define ASYNC_OK 0
#endif

typedef int v4i_t __attribute__((ext_vector_type(4)));
#if ASYNC_OK
typedef __attribute__((address_space(1))) v4i_t* g_v4i_ptr;   // global
typedef __attribute__((address_space(3))) v4i_t* l_v4i_ptr;   // LDS
#endif

#if ASYNC_OK
#if __has_builtin(__builtin_amdgcn_s_wait_asynccnt)
#define WAITA(K) __builtin_amdgcn_s_wait_asynccnt(K)
#else
#define WAITA(K) asm volatile("s_wait_asynccnt " #K ::: "memory")
#endif
#else
#define WAITA(K) ((void)0)
#endif

__device__ __forceinline__ void async_copy16(const float4* g, float4* l) {
#if ASYNC_OK
  __builtin_amdgcn_global_load_async_to_lds_b128(
      (g_v4i_ptr)(uintptr_t)g,   // global src  (as1 v4i*)
      (l_v4i_ptr)l,              // LDS dst     (as3 v4i*, addrspacecast)
      /*offset=*/0, /*cpol=*/0);
#else
  *l = *g;  // fallback: synchronous copy through VGPRs
#endif
}

// Compile-time-constant wait: k is a constant after full unroll, so the
// switch folds to a single s_wait_asynccnt immediate.
__device__ __forceinline__ void wait_async_upto(int k) {
#if ASYNC_OK
  switch (k) {
    case 0: WAITA(0); break;
    case 1: WAITA(1); break;
    case 2: WAITA(2); break;
    case 3: WAITA(3); break;
    case 4: WAITA(4); break;
    case 5: WAITA(5); break;
    case 6: WAITA(6); break;
    default: WAITA(7); break;
  }
  asm volatile("" ::: "memory");  // keep the LDS read below the wait
#endif
}

__device__ __forceinline__ float fast_exp2(float v) {
#if defined(__HIP_DEVICE_COMPILE__)
  return __builtin_amdgcn_exp2f(v);   // v_exp_f32 (trans)
#else
  return v;
#endif
}

// Per-component polynomial coefficients in log2-space + pre-exp'ed weights.
struct GmmParams {
  float ka[4], kb[4], kc[4], k0[4], k1[4], kk[4], W0[4], W1[4];
};

__device__ __forceinline__ GmmParams load_params(const float* __restrict__ means,
                                                 const float* __restrict__ ivc,
                                                 const float* __restrict__ lw) {
  GmmParams p;
  const float L2E = 1.4426950408889634f;
  const float S   = -0.5f * L2E;  // fold -0.5 and ln->log2 into coeffs
#pragma unroll
  for (int k = 0; k < 4; ++k) {
    float a = ivc[k*4+0], b01 = ivc[k*4+1], b10 = ivc[k*4+2], c = ivc[k*4+3];
    float B = b01 + b10;
    float m0 = means[k*2+0], m1 = means[k*2+1];
    p.ka[k] = S * a;
    p.kb[k] = S * B;
    p.kc[k] = S * c;
    p.k0[k] = S * (-2.0f*a*m0 - B*m1);
    p.k1[k] = S * (-2.0f*c*m1 - B*m0);
    p.kk[k] = S * (a*m0*m0 + B*m0*m1 + c*m1*m1);
    p.W0[k] = fast_exp2(lw[k*2+0] * L2E);  // exp(log_weights)
    p.W1[k] = fast_exp2(lw[k*2+1] * L2E);
  }
  return p;
}

// Returns the (mult0, mult1) multipliers exp(lse) for one channel pair.
__device__ __forceinline__ float2 gmm_pair(float x0, float x1, const GmmParams& p) {
  float x00 = x0*x0, x01 = x0*x1, x11 = x1*x1;
  float g0 = fmaf(p.ka[0],x00, fmaf(p.kb[0],x01, fmaf(p.kc[0],x11, fmaf(p.k0[0],x0, fmaf(p.k1[0],x1, p.kk[0])))));
  float g1 = fmaf(p.ka[1],x00, fmaf(p.kb[1],x01, fmaf(p.kc[1],x11, fmaf(p.k0[1],x0, fmaf(p.k1[1],x1, p.kk[1])))));
  float g2 = fmaf(p.ka[2],x00, fmaf(p.kb[2],x01, fmaf(p.kc[2],x11, fmaf(p.k0[2],x0, fmaf(p.k1[2],x1, p.kk[2])))));
  float g3 = fmaf(p.ka[3],x00, fmaf(p.kb[3],x01, fmaf(p.kc[3],x11, fmaf(p.k0[3],x0, fmaf(p.k1[3],x1, p.kk[3])))));
  float m  = fmaxf(fmaxf(g0, g1), fmaxf(g2, g3));
  float e0 = fast_exp2(g0 - m);
  float e1 = fast_exp2(g1 - m);
  float e2 = fast_exp2(g2 - m);
  float e3 = fast_exp2(g3 - m);
  float s0 = fmaf(e3, p.W0[3], fmaf(e2, p.W0[2], fmaf(e1, p.W0[1], e0 * p.W0[0])));
  float s1 = fmaf(e3, p.W1[3], fmaf(e2, p.W1[2], fmaf(e1, p.W1[1], e0 * p.W1[0])));
  float em = fast_exp2(m);
  return make_float2(em * s0, em * s1);
}

__device__ __forceinline__ float4 gmm_vec4(float4 v, const GmmParams& p) {
  float2 mA = gmm_pair(v.x, v.y, p);
  float2 mB = gmm_pair(v.z, v.w, p);
  float4 r;
  r.x = v.x * mA.x; r.y = v.y * mA.y;
  r.z = v.z * mB.x; r.w = v.w * mB.y;
  return r;
}

// ---- fast path: n divisible by TPB*4*DEPTH; each block owns DEPTH tiles ----
__global__ void __launch_bounds__(TPB)
gmm2d_act_async_kernel(const float* __restrict__ x,
                       const float* __restrict__ means,
                       const float* __restrict__ ivc,
                       const float* __restrict__ lw,
                       float* __restrict__ out) {
  __shared__ float4 buf[DEPTH][TPB];
  const GmmParams p = load_params(means, ivc, lw);  // uniform -> scalar loads
  const int tid  = threadIdx.x;
  const int base = blockIdx.x * (DEPTH * TPB) + tid;   // float4 units, 32-bit
  const float4* __restrict__ x4 = (const float4*)x;
  float4* __restrict__ o4 = (float4*)out;

  // burst-issue the whole block's input as DEPTH async b128 loads per lane
#pragma unroll
  for (int s = 0; s < DEPTH; ++s)
    async_copy16(&x4[base + s * TPB], &buf[s][tid]);

  // fully unrolled consume: stage s is resident once ASYNCcnt <= DEPTH-1-s
#pragma unroll
  for (int s = 0; s < DEPTH; ++s) {
    wait_async_upto(DEPTH - 1 - s);
    float4 v = buf[s][tid];               // ds_load_b128, const offset
    o4[base + s * TPB] = gmm_vec4(v, p);  // global_store_b128
  }
}

// ---- generic fallback: any n, plain vectorized grid-stride ----
__global__ void __launch_bounds__(TPB)
gmm2d_act_generic_kernel(const float* __restrict__ x,
                         const float* __restrict__ means,
                         const float* __restrict__ ivc,
                         const float* __restrict__ lw,
                         float* __restrict__ out, long long n) {
  const GmmParams p = load_params(means, ivc, lw);
  const long long n4 = n >> 2;
  const long long gstride = (long long)gridDim.x * TPB;
  const float4* __restrict__ x4 = (const float4*)x;
  float4* __restrict__ o4 = (float4*)out;
  for (long long i = (long long)blockIdx.x * TPB + threadIdx.x; i < n4; i += gstride)
    o4[i] = gmm_vec4(x4[i], p);
  for (long long i = (n4 << 2) + (long long)blockIdx.x * TPB + threadIdx.x; i < n;
       i += gstride) {
    long long pb = i & ~1LL;
    float2 mm = gmm_pair(x[pb], x[pb + 1], p);
    out[i] = x[i] * ((i & 1) ? mm.y : mm.x);
  }
}

extern "C" void kernel_launch(void* const* d_in, const int* in_sizes, int n_in,
                              void* d_out, int out_size, void* d_ws, size_t ws_size,
                              hipStream_t stream) {
  const float* x     = (const float*)d_in[0];  // (2,4096,2048) f32
  const float* means = (const float*)d_in[1];  // (1,1,4,2)
  const float* ivc   = (const float*)d_in[2];  // (4,2,2)
  const float* lw    = (const float*)d_in[3];  // (1,1,1,4,2)
  float* out = (float*)d_out;
  long long n = (long long)in_sizes[0];

  const long long per_block = (long long)TPB * 4 * DEPTH;   // 8192 elements
  if (n > 0 && (n % per_block) == 0) {
    int blocks = (int)(n / per_block);                      // 2048 here
    gmm2d_act_async_kernel<<<blocks, TPB, 0, stream>>>(x, means, ivc, lw, out);
  } else {
    long long n4 = n >> 2;
    long long b = (n4 + TPB - 1) / TPB;
    if (b < 1) b = 1;
    if (b > 2048) b = 2048;
    gmm2d_act_generic_kernel<<<(int)b, TPB, 0, stream>>>(x, means, ivc, lw, out, n);
  }
}